// BinaryMemory_67345087201876
// MI455X (gfx1250) — compile-verified
//
#include <hip/hip_runtime.h>
#include <math.h>

// ---------------------------------------------------------------------------
// BinaryMemory retrieve on MI455X (gfx1250, wave32)
//
// Phase 1: gated = sigmoid(query @ W.T + b)       [1 x 4096]   -- WMMA GEMV
// Phase 2: sims[n] = 1 - mean(|memory[n]-gated|)  [16384]      -- streaming L1
//          mask[n] = sims[n] >= 0.8
//
// Both phases are HBM-bandwidth bound (64 MB + 256 MB @ 23.3 TB/s ~= 14 us).
// ---------------------------------------------------------------------------

typedef __attribute__((ext_vector_type(2))) float v2f;
typedef __attribute__((ext_vector_type(8))) float v8f;

// ---------------------------------------------------------------------------
// Phase 1: one block (8 waves) per 16 output columns d0..d0+15.
// Each wave owns a K-slice of 512 -> 128 V_WMMA_F32_16X16X4_F32 ops, split
// across 4 independent accumulators to break the WMMA->WMMA RAW chain and
// keep 8 global loads in flight per loop iteration.
//   A (16x4 f32): row 0 = query[k..k+3], rows 1..15 zero-padded.
//     Layout: VGPR0 = K0 (lanes 0-15) / K2 (lanes 16-31); VGPR1 = K1 / K3.
//   B (4x16 f32): B[k][n] = W[d0+n][kbase+k]  (out[d] = dot(q, W[d,:])).
//     Layout mirrors C/D striping: VGPR0 = K0/K2 rows, VGPR1 = K1/K3 rows.
//   C/D (16x16 f32, 8 VGPRs): row M=0 sits in VGPR0 of lanes 0-15.
// ---------------------------------------------------------------------------
__global__ __launch_bounds__(256) void gated_gemv_wmma(
    const float* __restrict__ q, const float* __restrict__ W,
    const float* __restrict__ bias, float* __restrict__ gated, int D)
{
    __shared__ float red[8 * 16];          // per-wave row-0 partials

    const int lane = threadIdx.x & 31;
    const int wv   = threadIdx.x >> 5;     // 0..7
    const int d0   = blockIdx.x << 4;      // 16 outputs per block
    const int half = lane >> 4;            // 0: K0/K1 side, 1: K2/K3 side

    // Row-0-only mask for the A operand (avoids EXEC divergence at the WMMA).
    const float amask = ((lane & 15) == 0) ? 1.0f : 0.0f;

    // Each lane streams one W row contiguously (8B per iter, stride 16B;
    // lane n and lane n+16 jointly consume row d0+n back-to-back).
    const float* __restrict__ brow =
        W + (size_t)(d0 + (lane & 15)) * (size_t)D + (half << 1);
    const float* __restrict__ qh = q + (half << 1);

    const int kslice = D >> 3;             // 512 K per wave
    const int kbeg   = wv * kslice;
    const int kend   = kbeg + kslice;

    v8f c0 = {}, c1 = {}, c2 = {}, c3 = {};
    for (int k = kbeg; k < kend; k += 16) {
        v2f a0 = *(const v2f*)(qh + k);
        v2f a1 = *(const v2f*)(qh + k + 4);
        v2f a2 = *(const v2f*)(qh + k + 8);
        v2f a3 = *(const v2f*)(qh + k + 12);
        v2f b0 = *(const v2f*)(brow + k);
        v2f b1 = *(const v2f*)(brow + k + 4);
        v2f b2 = *(const v2f*)(brow + k + 8);
        v2f b3 = *(const v2f*)(brow + k + 12);
        a0.x *= amask; a0.y *= amask;
        a1.x *= amask; a1.y *= amask;
        a2.x *= amask; a2.y *= amask;
        a3.x *= amask; a3.y *= amask;
        // (neg_a, A, neg_b, B, c_mod, C, reuse_a, reuse_b)
        c0 = __builtin_amdgcn_wmma_f32_16x16x4_f32(false, a0, false, b0,
                                                   (short)0, c0, false, false);
        c1 = __builtin_amdgcn_wmma_f32_16x16x4_f32(false, a1, false, b1,
                                                   (short)0, c1, false, false);
        c2 = __builtin_amdgcn_wmma_f32_16x16x4_f32(false, a2, false, b2,
                                                   (short)0, c2, false, false);
        c3 = __builtin_amdgcn_wmma_f32_16x16x4_f32(false, a3, false, b3,
                                                   (short)0, c3, false, false);
    }

    // Row M=0 of the C/D tile: VGPR0 (element 0) on lanes 0..15, N = lane.
    if (lane < 16) red[wv * 16 + lane] = c0[0] + c1[0] + c2[0] + c3[0];
    __syncthreads();

    if (threadIdx.x < 16) {
        float acc = 0.0f;
        #pragma unroll
        for (int w = 0; w < 8; ++w) acc += red[w * 16 + threadIdx.x];
        acc += bias[d0 + threadIdx.x];
        gated[d0 + threadIdx.x] = 1.0f / (1.0f + __expf(-acc));
    }
}

// ---------------------------------------------------------------------------
// Phase 2: one wave per stored pattern row. gated staged in LDS (16 KB).
// float4 loads -> 512B per wave-load, perfectly coalesced HBM streaming.
// Unrolled x2 with independent accumulators for more loads in flight.
// ---------------------------------------------------------------------------
__global__ __launch_bounds__(256) void l1_sims(
    const float* __restrict__ memory, const float* __restrict__ gated,
    float* __restrict__ sims, float* __restrict__ mask, int N, int D)
{
    __shared__ float g[4096];
    for (int i = threadIdx.x; i < D; i += blockDim.x) g[i] = gated[i];
    __syncthreads();

    const int lane = threadIdx.x & 31;
    const int wv   = threadIdx.x >> 5;
    const int n    = blockIdx.x * 8 + wv;
    if (n >= N) return;

    const float4* __restrict__ mrow =
        (const float4*)(memory + (size_t)n * (size_t)D);
    const float4* __restrict__ grow = (const float4*)g;

    const int nvec = D >> 2;               // 1024 float4's per row
    float s0 = 0.0f, s1 = 0.0f;
    for (int i = lane; i < nvec; i += 64) {
        float4 ma = mrow[i];
        float4 mb = mrow[i + 32];
        float4 ga = grow[i];
        float4 gb = grow[i + 32];
        s0 += fabsf(ma.x - ga.x) + fabsf(ma.y - ga.y) +
              fabsf(ma.z - ga.z) + fabsf(ma.w - ga.w);
        s1 += fabsf(mb.x - gb.x) + fabsf(mb.y - gb.y) +
              fabsf(mb.z - gb.z) + fabsf(mb.w - gb.w);
    }
    float s = s0 + s1;

    // wave32 butterfly reduction
    #pragma unroll
    for (int off = 16; off > 0; off >>= 1) s += __shfl_xor(s, off, 32);

    if (lane == 0) {
        float sim = 1.0f - s / (float)D;
        sims[n] = sim;
        if (mask) mask[n] = (sim >= 0.8f) ? 1.0f : 0.0f;
    }
}

// ---------------------------------------------------------------------------
// Inputs (setup_inputs order): query[1,D], W[D,D], b[D], memory[N,D], all f32.
// Output: sims[N] f32 followed by mask[N] (0/1 as f32), concatenated flat.
// d_ws: gated[D] f32 scratch.
// ---------------------------------------------------------------------------
extern "C" void kernel_launch(void* const* d_in, const int* in_sizes, int n_in,
                              void* d_out, int out_size, void* d_ws, size_t ws_size,
                              hipStream_t stream)
{
    const float* q      = (const float*)d_in[0];
    const float* W      = (const float*)d_in[1];
    const float* bias   = (const float*)d_in[2];
    const float* memory = (const float*)d_in[3];

    const int D = in_sizes[0];          // 4096
    const int N = in_sizes[3] / D;      // 16384

    float* gated = (float*)d_ws;        // D floats of scratch
    float* sims  = (float*)d_out;
    float* mask  = (out_size >= 2 * N) ? (sims + N) : nullptr;

    // Phase 1: 256 blocks x 256 threads; block b computes gated[16b..16b+15].
    gated_gemv_wmma<<<D / 16, 256, 0, stream>>>(q, W, bias, gated, D);

    // Phase 2: 8 rows per block (one per wave), 2048 blocks.
    l1_sims<<<(N + 7) / 8, 256, 0, stream>>>(memory, gated, sims, mask, N, D);
}